// CrossAttentionBlock_78881369358733
// MI455X (gfx1250) — compile-verified
//
#include <hip/hip_runtime.h>
#include <hip/hip_bf16.h>
#include <math.h>

// Problem constants (from reference)
#define BB   8
#define NN   1024
#define CC   256
#define PD   16
#define EPSV 1e-5f

typedef __attribute__((ext_vector_type(2))) float v2f;
typedef __attribute__((ext_vector_type(8))) float v8f;

// ---------------------------------------------------------------------------
// Kernel 1: the entire "small" chain in one workgroup (8 waves, wave32).
//   ctx = param_tokens @ Wparam + bparam          (8x16 . 16x256, VALU)
//   ctx = layernorm(ctx)                           (per-row wave reduction)
//   v   = ctx @ Wkv[:, C:2C]                       (WMMA f32 16x16x4, K=256)
//   w   = v @ Wout + bout                          (WMMA f32 16x16x4, K=256)
// w (8x256, padded to 16x256) goes to workspace for the broadcast-add kernel.
// ---------------------------------------------------------------------------
__global__ __launch_bounds__(256) void fused_param_chain_kernel(
    const float* __restrict__ param_tokens,  // [8,16]
    const float* __restrict__ Wparam,        // [16,256]
    const float* __restrict__ bparam,        // [256]
    const float* __restrict__ ctx_g,         // [256]
    const float* __restrict__ ctx_b,         // [256]
    const float* __restrict__ Wkv,           // [256,512]
    const float* __restrict__ Wout,          // [256,256]
    const float* __restrict__ bout,          // [256]
    float* __restrict__ w_out)               // [16,256] scratch (rows 0..7 valid)
{
    __shared__ float ctx_s[16][CC];   // layernormed ctx, rows 8..15 = 0 (M padding)
    __shared__ float v_s[16][CC];     // v, rows 8..15 = 0 via zero-padded A

    const int tid  = threadIdx.x;     // 0..255
    const int wave = tid >> 5;        // 0..7
    const int lane = tid & 31;        // wave32

    // ---- Stage A: ctx = param_tokens @ Wparam + bparam (one column per thread)
    {
        const int c = tid;
        float wp[PD];
#pragma unroll
        for (int p = 0; p < PD; ++p) wp[p] = Wparam[p * CC + c];
        const float bp = bparam[c];
#pragma unroll
        for (int b = 0; b < BB; ++b) {
            float acc = bp;
#pragma unroll
            for (int p = 0; p < PD; ++p) acc += param_tokens[b * PD + p] * wp[p];
            ctx_s[b][c] = acc;
        }
#pragma unroll
        for (int b = BB; b < 16; ++b) ctx_s[b][c] = 0.0f;  // M padding rows
    }
    __syncthreads();

    // ---- Stage B: layernorm row `wave` in place (8 waves <-> 8 rows), wave32 shfl
    {
        const int r = wave;
        float s = 0.0f;
#pragma unroll
        for (int i = 0; i < CC / 32; ++i) s += ctx_s[r][lane + 32 * i];
#pragma unroll
        for (int off = 16; off > 0; off >>= 1) s += __shfl_xor(s, off, 32);
        const float mean = s * (1.0f / CC);

        float vs = 0.0f;
#pragma unroll
        for (int i = 0; i < CC / 32; ++i) {
            float d = ctx_s[r][lane + 32 * i] - mean;
            vs += d * d;
        }
#pragma unroll
        for (int off = 16; off > 0; off >>= 1) vs += __shfl_xor(vs, off, 32);
        const float rstd = rsqrtf(vs * (1.0f / CC) + EPSV);

#pragma unroll
        for (int i = 0; i < CC / 32; ++i) {
            const int idx = lane + 32 * i;
            const float x = ctx_s[r][idx];
            ctx_s[r][idx] = (x - mean) * rstd * ctx_g[idx] + ctx_b[idx];
        }
    }
    __syncthreads();

    // WMMA fragment coordinates (ISA 7.12.2, 32-bit A 16x4 / C-D 16x16 layouts)
    const int rc = lane & 15;          // A row / B,D column within tile
    const int kh = (lane >> 4) * 2;    // K sub-offset: lanes 0-15 -> K0,K1; 16-31 -> K2,K3
    const int dR = 8 * (lane >> 4);    // D rows: lanes 0-15 -> 0..7, lanes 16-31 -> 8..15

    // ---- Stage C: v = ctx_ln @ Wkv[:, CC:2*CC]; each wave does 2 column tiles
#pragma unroll
    for (int t = 0; t < 2; ++t) {
        const int n0 = (wave * 2 + t) * 16;
        v8f acc = {};
        for (int k0 = 0; k0 < CC; k0 += 4) {
            v2f a, b;
            a.x = ctx_s[rc][k0 + kh];
            a.y = ctx_s[rc][k0 + kh + 1];
            b.x = Wkv[(k0 + kh)     * (2 * CC) + CC + n0 + rc];
            b.y = Wkv[(k0 + kh + 1) * (2 * CC) + CC + n0 + rc];
            acc = __builtin_amdgcn_wmma_f32_16x16x4_f32(
                /*neg_a=*/false, a, /*neg_b=*/false, b,
                /*c_mod=*/(short)0, acc, /*reuse_a=*/false, /*reuse_b=*/false);
        }
#pragma unroll
        for (int r = 0; r < 8; ++r) v_s[r + dR][n0 + rc] = acc[r];  // rows 8..15 = 0
    }
    __syncthreads();

    // ---- Stage D: w = v @ Wout + bout
#pragma unroll
    for (int t = 0; t < 2; ++t) {
        const int n0 = (wave * 2 + t) * 16;
        v8f acc = {};
        for (int k0 = 0; k0 < CC; k0 += 4) {
            v2f a, b;
            a.x = v_s[rc][k0 + kh];
            a.y = v_s[rc][k0 + kh + 1];
            b.x = Wout[(k0 + kh)     * CC + n0 + rc];
            b.y = Wout[(k0 + kh + 1) * CC + n0 + rc];
            acc = __builtin_amdgcn_wmma_f32_16x16x4_f32(
                false, a, false, b, (short)0, acc, false, false);
        }
        const float bo = bout[n0 + rc];
#pragma unroll
        for (int r = 0; r < 8; ++r) w_out[(r + dR) * CC + n0 + rc] = acc[r] + bo;
    }
}

// ---------------------------------------------------------------------------
// Kernel 2: out[b,n,:] = img_tokens[b,n,:] + w[b,:]   (pure HBM bandwidth)
// 16 MB total traffic -> ~0.7 us at 23.3 TB/s. float4 (b128) loads/stores.
// ---------------------------------------------------------------------------
__global__ __launch_bounds__(256) void broadcast_residual_add_kernel(
    const float* __restrict__ img,   // [8,1024,256]
    const float* __restrict__ w,     // [16,256] (rows 0..7 used)
    float* __restrict__ out)         // [8,1024,256]
{
    const long gid  = (long)blockIdx.x * blockDim.x + threadIdx.x; // 0..524287
    const long base = gid * 4;
    const int  b    = (int)(base >> 18);        // / (N*C) = / 2^18
    const int  c    = (int)(base & (CC - 1));   // column within C (float4 aligned)

    const float4 iv = *reinterpret_cast<const float4*>(img + base);
    const float4 wv = *reinterpret_cast<const float4*>(w + b * CC + c);
    float4 ov;
    ov.x = iv.x + wv.x;
    ov.y = iv.y + wv.y;
    ov.z = iv.z + wv.z;
    ov.w = iv.w + wv.w;
    *reinterpret_cast<float4*>(out + base) = ov;
}

// ---------------------------------------------------------------------------
// Launch. Input order (setup_inputs): 0 img_tokens, 1 param_tokens,
// 2 img_norm_g, 3 img_norm_b, 4 Wq, 5 Wparam, 6 bparam, 7 ctx_norm_g,
// 8 ctx_norm_b, 9 Wkv, 10 Wout, 11 bout.  (img_norm_*, Wq are dead code.)
// ---------------------------------------------------------------------------
extern "C" void kernel_launch(void* const* d_in, const int* in_sizes, int n_in,
                              void* d_out, int out_size, void* d_ws, size_t ws_size,
                              hipStream_t stream) {
    const float* img          = (const float*)d_in[0];
    const float* param_tokens = (const float*)d_in[1];
    const float* Wparam       = (const float*)d_in[5];
    const float* bparam       = (const float*)d_in[6];
    const float* ctx_g        = (const float*)d_in[7];
    const float* ctx_b        = (const float*)d_in[8];
    const float* Wkv          = (const float*)d_in[9];
    const float* Wout         = (const float*)d_in[10];
    const float* bout         = (const float*)d_in[11];
    float* out = (float*)d_out;
    float* w_scratch = (float*)d_ws;  // 16*256 floats

    fused_param_chain_kernel<<<1, 256, 0, stream>>>(
        param_tokens, Wparam, bparam, ctx_g, ctx_b, Wkv, Wout, bout, w_scratch);

    const int total4 = (BB * NN * CC) / 4;   // 524288
    broadcast_residual_add_kernel<<<total4 / 256, 256, 0, stream>>>(
        img, w_scratch, out);
}